// SinkhornProjection_60962765799638
// MI455X (gfx1250) — compile-verified
//
#include <hip/hip_runtime.h>
#include <hip/hip_bf16.h>

// Sinkhorn-Knopp on 65536 independent 32x32 fp32 matrices for MI455X (gfx1250).
//
// Strategy: keep K = exp(H/tau) resident in VGPRs in WMMA B-operand layout
// (plus its transpose), and run the 10 Sinkhorn iterations in scaling-vector
// (u,v) form:
//     u = 1/(K v + eps) ; v = 1/(K^T u + eps)
// Each matvec is done with V_WMMA_F32_16X16X4_F32: A = broadcast tiles of the
// vector (built with v_readlane -> SGPR broadcast, no LDS traffic), B =
// K-slices of the resident matrix, accumulated over 8 K=4 slices per 16-column
// half. The result lands per-lane in the C layout, so rcp and the next
// A-operand build are cheap. Final output = diag(u) K diag(v), written with
// non-temporal stores (write-once 256MB stream; don't displace L2 input lines).

typedef __attribute__((ext_vector_type(2))) float v2f;
typedef __attribute__((ext_vector_type(8))) float v8f;

#define TAU 0.1f
#define NUM_ITERS 10
#define EPS 1e-8f
// exp(x/TAU) = exp2(x * log2(e)/TAU)
#define EXP_SCALE (1.4426950408889634f / TAU)

// Uniform-index lane broadcast via v_readlane_b32 (SGPR result; avoids the
// ds_bpermute lowering of __shfl and its dscnt stalls).
__device__ __forceinline__ float bcast(float x, int k) {
  return __uint_as_float(__builtin_amdgcn_readlane(__float_as_uint(x), k));
}

// B-operand storage convention for a 32x32 matrix X, per column-tile j (0/1):
//   reg t = 2*s + p  (s = K-slice 0..7, p = 0/1)
//   lane L holds X[4*s + p + 2*(L>>4)][16*j + (L&15)]
// i.e. reg pair (2s, 2s+1) is exactly the WMMA B operand for rows 4s..4s+3.

// y = X^T w for one 16-wide half j of y (y[16j+n] returned at lane n&15).
// w is given as (w0, w1): w0 lane l = w[l&15], w1 lane l = w[16 + (l&15)].
__device__ __forceinline__ v8f matvec_half(const float (&B)[2][16], int j,
                                           float w0, float w1, int hi) {
  v8f acc = {};
#pragma unroll
  for (int s = 0; s < 8; ++s) {
    // Build A (16x4 broadcast of w[4s..4s+3]):
    //   vgpr0: lanes0-15 = w[4s],   lanes16-31 = w[4s+2]
    //   vgpr1: lanes0-15 = w[4s+1], lanes16-31 = w[4s+3]
    float ws = (s < 4) ? w0 : w1;
    int k0 = (4 * s) & 15;
    float e0 = bcast(ws, k0 + 0);
    float e1 = bcast(ws, k0 + 1);
    float e2 = bcast(ws, k0 + 2);
    float e3 = bcast(ws, k0 + 3);
    v2f A;
    A.x = hi ? e2 : e0;
    A.y = hi ? e3 : e1;
    v2f Bv;
    Bv.x = B[j][2 * s + 0];
    Bv.y = B[j][2 * s + 1];
    acc = __builtin_amdgcn_wmma_f32_16x16x4_f32(false, A, false, Bv,
                                                (short)0, acc, false, false);
  }
  return acc;
}

__global__ __launch_bounds__(256) void sinkhorn_wmma_kernel(
    const float* __restrict__ H, float* __restrict__ out, int nmat) {
  const int lane = threadIdx.x & 31;
  const int wave = threadIdx.x >> 5;                 // 8 waves per block
  const int h    = (lane >> 4) & 1;                  // lane-half select
  const int wavesPerGrid = gridDim.x * 8;

  for (int mat = blockIdx.x * 8 + wave; mat < nmat; mat += wavesPerGrid) {
    const float* __restrict__ Hm = H + (size_t)mat * 1024;
    float* __restrict__ Om = out + (size_t)mat * 1024;

    // Prefetch next matrix (global_prefetch_b8): 32 lanes * 128B stride = 4KB.
    if (mat + wavesPerGrid < nmat) {
      __builtin_prefetch(H + (size_t)(mat + wavesPerGrid) * 1024 + lane * 32,
                         0, 0);
    }

    // ---- Load K = max(exp(H/tau), eps) into B-layout regs, and K^T too ----
    float bK[2][16];   // K   in B layout (used for K^T u)
    float bT[2][16];   // K^T in B layout (used for K v)
#pragma unroll
    for (int t = 0; t < 16; ++t) {
      int s = t >> 1, p = t & 1;
      int row = 4 * s + p + 2 * h;
      int col = lane & 15;
      float x0 = Hm[row * 32 + col];
      float x1 = Hm[row * 32 + 16 + col];
      bK[0][t] = fmaxf(exp2f(x0 * EXP_SCALE), EPS);
      bK[1][t] = fmaxf(exp2f(x1 * EXP_SCALE), EPS);
      // K^T[row][col] = K[col][row]; second read hits L0 (same 4KB tile).
      float t0 = Hm[col * 32 + row];
      float t1 = Hm[(16 + col) * 32 + row];
      bT[0][t] = fmaxf(exp2f(t0 * EXP_SCALE), EPS);
      bT[1][t] = fmaxf(exp2f(t1 * EXP_SCALE), EPS);
    }

    // ---- Sinkhorn iterations in u/v form ----
    float u0, u1;
    float v0 = 1.0f, v1 = 1.0f;    // v starts as ones (first step = row norm)
#pragma unroll 1
    for (int it = 0; it < NUM_ITERS; ++it) {
      // row phase: u = 1/(K v + eps).  K v = (K^T)^T v -> use bT as B operand.
      v8f pu0 = matvec_half(bT, 0, v0, v1, h);
      v8f pu1 = matvec_half(bT, 1, v0, v1, h);
      u0 = __builtin_amdgcn_rcpf(pu0[0] + EPS);   // u[lane&15]      (rows 0-15)
      u1 = __builtin_amdgcn_rcpf(pu1[0] + EPS);   // u[16+(lane&15)] (rows 16-31)
      // col phase: v = 1/(K^T u + eps) -> use bK as B operand.
      v8f pv0 = matvec_half(bK, 0, u0, u1, h);
      v8f pv1 = matvec_half(bK, 1, u0, u1, h);
      v0 = __builtin_amdgcn_rcpf(pv0[0] + EPS);
      v1 = __builtin_amdgcn_rcpf(pv1[0] + EPS);
    }

    // ---- Output: out[m][n] = u[m] * K[m][n] * v[n] (non-temporal stores) ----
#pragma unroll
    for (int t = 0; t < 16; ++t) {
      int s = t >> 1, p = t & 1;
      int r0 = 4 * s + p;                       // low-half row; hi-half = r0+2
      float uw = (s < 4) ? u0 : u1;             // rows 4s..4s+3 share a u reg
      float ua = bcast(uw, (r0 + 0) & 15);
      float ub = bcast(uw, (r0 + 2) & 15);
      float uval = h ? ub : ua;                 // u[row] for this lane's row
      int row = r0 + 2 * h;
      int col = lane & 15;
      __builtin_nontemporal_store(uval * bK[0][t] * v0, &Om[row * 32 + col]);
      __builtin_nontemporal_store(uval * bK[1][t] * v1, &Om[row * 32 + 16 + col]);
    }
  }
}

extern "C" void kernel_launch(void* const* d_in, const int* in_sizes, int n_in,
                              void* d_out, int out_size, void* d_ws, size_t ws_size,
                              hipStream_t stream) {
  (void)n_in; (void)out_size; (void)d_ws; (void)ws_size;
  const float* H = (const float*)d_in[0];
  float* out = (float*)d_out;
  int nmat = in_sizes[0] / 1024;            // 65536 matrices of 32x32
  int blocks = (nmat + 7) / 8;              // 8 waves (matrices) per block
  sinkhorn_wmma_kernel<<<blocks, 256, 0, stream>>>(H, out, nmat);
}